// TheirDensityRegressor_16810501996693
// MI455X (gfx1250) — compile-verified
//
#include <hip/hip_runtime.h>
#include <math.h>

// ---------------------------------------------------------------------------
// LSTM(H=8,F=5) + MLP(8->5->5->1), fused, fp32 WMMA on gfx1250.
// One wave (32 lanes) owns 16 batch rows for all T steps.
// - Recurrent + input GEMMs: V_WMMA_F32_16X16X4_F32 (exact fp32, 100% MAC use),
//   two independent 4-deep accumulate chains (ILP 2).
// - x staged global->LDS one chunk ahead via GLOBAL_LOAD_ASYNC_TO_LDS_B128
//   (ASYNCcnt), waited with s_wait_asynccnt at the chunk boundary only.
// - h re-laid-out D->A through LDS each step (DS ops are wave-ordered).
// - Gate nonlinearities reformulated as pure V_TANH_F32 trans ops.
// - MLP deferred per 32-step chunk, full-wave parallel, contiguous out stores.
// ---------------------------------------------------------------------------

typedef __attribute__((ext_vector_type(2))) float v2f;
typedef __attribute__((ext_vector_type(8))) float v8f;

#define LSTM_CHUNK 32

__device__ __forceinline__ float dtanh(float x) {
#if __has_builtin(__builtin_amdgcn_tanhf)
  return __builtin_amdgcn_tanhf(x);       // V_TANH_F32 trans op
#elif __has_builtin(__builtin_amdgcn_tanh_f32)
  return __builtin_amdgcn_tanh_f32(x);
#else
  return tanhf(x);
#endif
}

__device__ __forceinline__ v8f wmma4(v2f a, v2f b, v8f c) {
  // D = A(16x4) * B(4x16) + C(16x16), fp32
  return __builtin_amdgcn_wmma_f32_16x16x4_f32(false, a, false, b, (short)0, c,
                                               false, false);
}

__device__ __forceinline__ void wait_async0() {
#if __has_builtin(__builtin_amdgcn_s_wait_asynccnt)
  __builtin_amdgcn_s_wait_asynccnt(0);
#else
  asm volatile("s_wait_asynccnt 0x0" ::: "memory");
#endif
}

// Async DMA: 16 bytes global -> LDS, tracked by ASYNCcnt (no VGPR data path).
__device__ __forceinline__ void async_b128(unsigned lds_off, const void* gaddr) {
  asm volatile("global_load_async_to_lds_b128 %0, %1, off"
               :: "v"(lds_off), "v"(gaddr)
               : "memory");
}

__launch_bounds__(32)
__global__ void lstm_mlp_kernel(const float* __restrict__ x,
                                const float* __restrict__ W_ih,
                                const float* __restrict__ W_hh,
                                const float* __restrict__ b_ih,
                                const float* __restrict__ b_hh,
                                const float* __restrict__ W1,
                                const float* __restrict__ b1,
                                const float* __restrict__ W2,
                                const float* __restrict__ b2,
                                const float* __restrict__ W3,
                                const float* __restrict__ b3,
                                float* __restrict__ out, int T) {
  __shared__ float xs[2][16 * LSTM_CHUNK * 5];    // staged x, double buffered
  __shared__ float hbuf[LSTM_CHUNK * 16 * 8];     // h history (also A-transpose)

  const int lane = threadIdx.x;
  const int lm   = lane & 15;
  const bool hiH = lane >= 16;
  const int c0   = hiH ? 2 : 0;                   // K sub-column for this half
  const int row0 = blockIdx.x * 16;

  // ---- B-matrix-layout weights (K=4 x N=16): v0 holds K={0|2}, v1 K={1|3} ----
  v2f bw[2][2], bx[2][2];
  v8f cbias[2];
#pragma unroll
  for (int tile = 0; tile < 2; ++tile) {
    const int g = tile * 16 + lm;                 // gate/output column
#pragma unroll
    for (int ch = 0; ch < 2; ++ch) {              // K chunks: j 0..3 / 4..7
      const int j = ch * 4 + c0;
      bw[tile][ch].x = W_hh[g * 8 + j];
      bw[tile][ch].y = W_hh[g * 8 + j + 1];
    }
    bx[tile][0].x = W_ih[g * 5 + c0];
    bx[tile][0].y = W_ih[g * 5 + c0 + 1];
    bx[tile][1].x = hiH ? 0.0f : W_ih[g * 5 + 4]; // K=4 valid, K=5..7 zero
    bx[tile][1].y = 0.0f;
    const float bs = b_ih[g] + b_hh[g];
#pragma unroll
    for (int m = 0; m < 8; ++m) cbias[tile][m] = bs;  // bias as initial C
  }

  // ---- MLP weights (uniform loads, persistent) ----
  float w1[40], w2[25], w3v[5], b1v[5], b2v[5];
#pragma unroll
  for (int i = 0; i < 40; ++i) w1[i] = W1[i];
#pragma unroll
  for (int i = 0; i < 25; ++i) w2[i] = W2[i];
#pragma unroll
  for (int i = 0; i < 5; ++i) { w3v[i] = W3[i]; b1v[i] = b1[i]; b2v[i] = b2[i]; }
  const float b3s = b3[0];

  // ---- state ----
  v2f ah0 = {0.0f, 0.0f}, ah1 = {0.0f, 0.0f};     // h in A layout (starts at 0)
  float cst[8], hst[8];
#pragma unroll
  for (int m = 0; m < 8; ++m) { cst[m] = 0.0f; hst[m] = 0.0f; }

  // ---- async x chunk prefetch: global -> LDS, no VGPR staging ----
  const int TV = (T * 5) / 4;                     // float4 per batch row
  const float4* xsrc = (const float4*)x;
  auto prefetchX = [&](int cbk, int buf) {
    const unsigned ldsbase = (unsigned)(uintptr_t)(&xs[buf][0]);
#pragma unroll
    for (int k = 0; k < 20; ++k) {                // 640 b128 per chunk, 20 instrs
      const int idx = lane + 32 * k;
      const int r = idx / 40, q = idx % 40;       // 40 float4 per row-chunk
      const float4* g = xsrc + (size_t)(row0 + r) * TV + (size_t)cbk * 40 + q;
      async_b128(ldsbase + (unsigned)idx * 16u, (const void*)g);
    }
  };

  const int NCH = T / LSTM_CHUNK;
  prefetchX(0, 0);
  wait_async0();
  __builtin_amdgcn_wave_barrier();

  for (int cbk = 0; cbk < NCH; ++cbk) {
    const int buf = cbk & 1;
    if (cbk + 1 < NCH) prefetchX(cbk + 1, buf ^ 1);  // DMA next chunk now
    const float* xsb = xs[buf];
    const int tbase = cbk * LSTM_CHUNK;

#pragma unroll 1
    for (int tt = 0; tt < LSTM_CHUNK; ++tt) {
      // x_t in A layout (16x8, K=0..4 valid) -- unconditional loads + select
      const int xb = lm * (LSTM_CHUNK * 5) + tt * 5;
      const float x01a = xsb[xb + c0];
      const float x01b = xsb[xb + c0 + 1];
      const float x4   = xsb[xb + 4];
      v2f ax0, ax1;
      ax0.x = x01a;
      ax0.y = x01b;
      ax1.x = hiH ? 0.0f : x4;
      ax1.y = 0.0f;

      // gates = [x|h] @ [W_ih|W_hh]^T + bias : two 16x16 D tiles, 8 WMMAs
      v8f g0 = wmma4(ah0, bw[0][0],
               wmma4(ah1, bw[0][1],
               wmma4(ax0, bx[0][0],
               wmma4(ax1, bx[0][1], cbias[0]))));
      v8f g1 = wmma4(ah0, bw[1][0],
               wmma4(ah1, bw[1][1],
               wmma4(ax0, bx[1][0],
               wmma4(ax1, bx[1][1], cbias[1]))));

      // activations: tile0 = sigmoid(i,f); tile1 = tanh(g) | sigmoid(o)
      const bool og = (lane & 8) != 0;            // N%16 >= 8 within tile
      float a0[8], a1[8];
#pragma unroll
      for (int m = 0; m < 8; ++m) {
        a0[m] = 0.5f + 0.5f * dtanh(0.5f * g0[m]);          // sigmoid
        const float v  = g1[m];
        const float tu = dtanh(og ? 0.5f * v : v);
        a1[m] = og ? 0.5f + 0.5f * tu : tu;
      }
      // state update; j-lanes ((lane&8)==0) hold hidden unit j = lane&7
#pragma unroll
      for (int m = 0; m < 8; ++m) {
        const float fm = __shfl_xor(a0[m], 8, 32);          // f from N=j+8
        const float om = __shfl_xor(a1[m], 8, 32);          // o from N=j+8
        cst[m] = fm * cst[m] + a0[m] * a1[m];               // f*c + i*g
        hst[m] = om * dtanh(cst[m]);
      }
      // h -> LDS (history slot tt), then reload in A layout for next step
      if ((lane & 8) == 0) {
        const int j = lane & 7;
        const int ro = hiH ? 8 : 0;
#pragma unroll
        for (int m = 0; m < 8; ++m) hbuf[tt * 128 + (m + ro) * 8 + j] = hst[m];
      }
      __builtin_amdgcn_wave_barrier();
      const int hb = tt * 128 + lm * 8 + c0;
      ah0.x = hbuf[hb];     ah0.y = hbuf[hb + 1];
      ah1.x = hbuf[hb + 4]; ah1.y = hbuf[hb + 5];
    }

    // ---- deferred MLP over the 32-step chunk (full-wave parallel) ----
    __builtin_amdgcn_wave_barrier();
    {
      const int r   = lane >> 1;                  // batch row 0..15
      const int tof = (lane & 1) * 16;            // each lane: 16 contiguous t
#pragma unroll 1
      for (int k = 0; k < 16; ++k) {
        const int tt = tof + k;
        const float* hp = &hbuf[tt * 128 + r * 8];
        float hv[8];
#pragma unroll
        for (int j = 0; j < 8; ++j) hv[j] = fmaxf(hp[j], 0.0f);   // relu(h)
        float z1[5];
#pragma unroll
        for (int kk = 0; kk < 5; ++kk) {
          float s = b1v[kk];
#pragma unroll
          for (int j = 0; j < 8; ++j) s = fmaf(w1[kk * 8 + j], hv[j], s);
          z1[kk] = fmaxf(s, 0.0f);
        }
        float z2[5];
#pragma unroll
        for (int kk = 0; kk < 5; ++kk) {
          float s = b2v[kk];
#pragma unroll
          for (int j = 0; j < 5; ++j) s = fmaf(w2[kk * 5 + j], z1[j], s);
          z2[kk] = fmaxf(s, 0.0f);
        }
        float o = b3s;
#pragma unroll
        for (int j = 0; j < 5; ++j) o = fmaf(w3v[j], z2[j], o);
        out[(size_t)(row0 + r) * T + tbase + tt] = o;
      }
    }

    if (cbk + 1 < NCH) {                          // next chunk's DMA must land
      wait_async0();
      __builtin_amdgcn_wave_barrier();
    }
  }
}

extern "C" void kernel_launch(void* const* d_in, const int* in_sizes, int n_in,
                              void* d_out, int out_size, void* d_ws, size_t ws_size,
                              hipStream_t stream) {
  const float* x    = (const float*)d_in[0];
  const float* W_ih = (const float*)d_in[1];
  const float* W_hh = (const float*)d_in[2];
  const float* b_ih = (const float*)d_in[3];
  const float* b_hh = (const float*)d_in[4];
  const float* W1   = (const float*)d_in[5];
  const float* b1   = (const float*)d_in[6];
  const float* W2   = (const float*)d_in[7];
  const float* b2   = (const float*)d_in[8];
  const float* W3   = (const float*)d_in[9];
  const float* b3   = (const float*)d_in[10];
  float* out = (float*)d_out;

  const int T = 1024;                             // fixed by the reference
  const int B = (in_sizes[0] / 5) / T;            // 4096
  dim3 grid(B / 16), block(32);
  hipLaunchKernelGGL(lstm_mlp_kernel, grid, block, 0, stream,
                     x, W_ih, W_hh, b_ih, b_hh, W1, b1, W2, b2, W3, b3, out, T);
}